// FinePreprocess_26345329393943
// MI455X (gfx1250) — compile-verified
//
#include <hip/hip_runtime.h>

typedef float v2f __attribute__((ext_vector_type(2)));
typedef float v8f __attribute__((ext_vector_type(8)));
typedef int   v4i __attribute__((ext_vector_type(4)));

#define CF    128          // fine channels
#define CC    256          // coarse channels
#define HF    480
#define WF    640
#define IMG   (HF * WF)
#define WC_   80
#define STR_  8
#define WW    25           // 5x5 window
#define BPAD  136          // Bs row pitch (16B aligned, conflict-free halves)
#define APAD  132          // As row pitch

#if __has_builtin(__builtin_amdgcn_global_load_async_to_lds_b128) && \
    __has_builtin(__builtin_amdgcn_global_load_async_to_lds_b32)
#define ASYNC_LDS 1
#else
#define ASYNC_LDS 0
#endif
// Probe result (round 4): device (gfx1250) pass has ASYNC_LDS == 1; only the
// host pass lacks the builtins. So the async-to-LDS path is active on device.

#if ASYNC_LDS
// Builtin signatures (confirmed by clang diagnostics):
//   b128: (v4i addrspace(1)*, v4i addrspace(3)*, imm int, imm int)
//   b32 : (int addrspace(1)*, int addrspace(3)*, imm int, imm int)
// Generic LDS pointer: low 32 bits hold the LDS byte offset (ISA aperture rule).
__device__ __forceinline__ void async_g2l_b128(const float* src, float* dst) {
  typedef __attribute__((address_space(1))) v4i gv4i;
  typedef __attribute__((address_space(3))) v4i lv4i;
  __builtin_amdgcn_global_load_async_to_lds_b128(
      (gv4i*)(uintptr_t)src,
      (lv4i*)(unsigned int)(uintptr_t)dst, 0, 0);
}
__device__ __forceinline__ void async_g2l_b32(const float* src, float* dst) {
  typedef __attribute__((address_space(1))) int gi;
  typedef __attribute__((address_space(3))) int li;
  __builtin_amdgcn_global_load_async_to_lds_b32(
      (gi*)(uintptr_t)src,
      (li*)(unsigned int)(uintptr_t)dst, 0, 0);
}
__device__ __forceinline__ void wait_async0() {
#if __has_builtin(__builtin_amdgcn_s_wait_asynccnt)
  __builtin_amdgcn_s_wait_asynccnt(0);
#else
  asm volatile("s_wait_asynccnt 0x0" ::: "memory");
#endif
}
#endif

static __device__ __forceinline__ v8f wmma_f32(v2f a, v2f b, v8f c) {
  // D(16x16,f32) = A(16x4,f32) x B(4x16,f32) + C
  return __builtin_amdgcn_wmma_f32_16x16x4_f32(
      false, a, false, b, (short)0, c, false, false);
}

// Kernel 2: out[row] = gathered_window_row(128) @ merge_w[0:128,:] + t[match(row)]
// row = (side*M + m)*25 + p ; 16 rows per block, 8 waves x 16-wide N tiles.
__global__ __launch_bounds__(256) void fine_kernel(
    const float* __restrict__ feat_f0, const float* __restrict__ feat_f1,
    const float* __restrict__ merge_w,
    const int* __restrict__ b_ids, const int* __restrict__ i_ids,
    const int* __restrict__ j_ids,
    const float* __restrict__ t_in, float* __restrict__ out, int M)
{
  __shared__ float As[16][APAD];
  __shared__ float Bs[CF][BPAD];
  __shared__ int   mmIdx[16];
  __shared__ int   sideArr[16];
  __shared__ int   offArr[16];     // element offset into feat map, -1 = padded

  const int tid  = threadIdx.x;
  const int lane = tid & 31;
  const int wave = tid >> 5;
  const int r0        = blockIdx.x * 16;
  const int totalRows = 2 * M * WW;     // <= ~150K, fits int easily

  if (tid < 16) {
    int r = r0 + tid;
    int mm = 0, side = 0, off = -1;
    if (r < totalRows) {
      side = (r >= M * WW) ? 1 : 0;
      int rem = r - side * M * WW;
      int m   = rem / WW;
      int p   = rem % WW;
      int idx = side ? j_ids[m] : i_ids[m];
      int b   = b_ids[m];
      int rr  = (idx / WC_) * STR_ + (p / 5) - 2;   // pad = 2
      int cc  = (idx % WC_) * STR_ + (p % 5) - 2;
      mm = side * M + m;
      if (rr >= 0 && rr < HF && cc >= 0 && cc < WF)
        off = b * CF * IMG + rr * WF + cc;          // < 2^27 elements
    }
    mmIdx[tid]   = mm;
    sideArr[tid] = side;
    offArr[tid]  = off;
  }
  __syncthreads();

  // ---- Stage B = merge_w[0:128][0:128] into LDS (coalesced 16B granules) ----
  #pragma unroll
  for (int j = 0; j < 16; ++j) {
    int g   = j * 256 + tid;          // float4 index 0..4095
    int row = g >> 5;                 // /32  -> K row
    int c4  = g & 31;                 // float4 within row
    const float* src = merge_w + g * 4;
    float* dst = &Bs[row][c4 * 4];
#if ASYNC_LDS
    async_g2l_b128(src, dst);
#else
    dst[0] = src[0]; dst[1] = src[1]; dst[2] = src[2]; dst[3] = src[3];
#endif
  }

  // ---- Gather A tile: row = tid&15, channels c = (tid>>4) + 16*j ----
  {
    const int row = tid & 15;
    const int cb  = tid >> 4;
    const int off = offArr[row];
    const float* fb = sideArr[row] ? feat_f1 : feat_f0;
    if (off >= 0) {
      #pragma unroll
      for (int j = 0; j < 8; ++j) {
        int c = cb + 16 * j;
        const float* src = fb + off + c * IMG;
        float* dst = &As[row][c];
#if ASYNC_LDS
        async_g2l_b32(src, dst);
#else
        dst[0] = src[0];
#endif
      }
    }
#if ASYNC_LDS
    wait_async0();
#endif
    if (off < 0) {
      #pragma unroll
      for (int j = 0; j < 8; ++j) As[row][cb + 16 * j] = 0.0f;
    }
  }
  __syncthreads();

  const int nl = lane & 15;
  const int kk = (lane >> 4) * 2;
  const int hi = lane >> 4;
  const int n0 = wave * 16;
  const int oc = n0 + nl;            // output column

  v8f acc;
  #pragma unroll
  for (int r = 0; r < 8; ++r)
    acc[r] = t_in[mmIdx[r + 8 * hi] * CF + oc];

  #pragma unroll 4
  for (int k0 = 0; k0 < CF; k0 += 4) {
    v2f a, b;
    a.x = As[nl][k0 + kk];
    a.y = As[nl][k0 + kk + 1];
    b.x = Bs[k0 + kk][oc - n0 + n0];  // == Bs[k0+kk][oc]
    b.y = Bs[k0 + kk + 1][oc];
    acc = wmma_f32(a, b, acc);
  }

  if (r0 + 16 <= totalRows) {        // uniform fast path: full tile
    #pragma unroll
    for (int r = 0; r < 8; ++r)
      out[(r0 + r + 8 * hi) * CF + oc] = acc[r];
  } else {
    #pragma unroll
    for (int r = 0; r < 8; ++r) {
      int rowG = r0 + r + 8 * hi;
      if (rowG < totalRows) out[rowG * CF + oc] = acc[r];
    }
  }
}

// Kernel 1: t[mm] = (coarse[mm] @ down_w + down_b) @ merge_w[128:256] + merge_b
__global__ __launch_bounds__(256) void prep_kernel(
    const float* __restrict__ feat_c0, const float* __restrict__ feat_c1,
    const float* __restrict__ down_w,  const float* __restrict__ down_b,
    const float* __restrict__ merge_w, const float* __restrict__ merge_b,
    const int* __restrict__ b_ids, const int* __restrict__ i_ids,
    const int* __restrict__ j_ids,
    float* __restrict__ t_out, int M, int LC)
{
  __shared__ float Cs[16][CC + 4];
  __shared__ float Ws[16][CF + 4];
  __shared__ const float* rowPtr[16];

  const int tid  = threadIdx.x;
  const int lane = tid & 31;
  const int wave = tid >> 5;
  const int mm0  = blockIdx.x * 16;
  const int twoM = 2 * M;

  if (tid < 16) {
    int mm = mm0 + tid;
    const float* p = nullptr;
    if (mm < twoM) {
      int s   = (mm >= M) ? 1 : 0;
      int m   = mm - s * M;
      int idx = s ? j_ids[m] : i_ids[m];
      int b   = b_ids[m];
      const float* base = s ? feat_c1 : feat_c0;
      p = base + (b * LC + idx) * CC;
    }
    rowPtr[tid] = p;
  }
  __syncthreads();

  #pragma unroll
  for (int j = 0; j < 16; ++j) {
    const float* p = rowPtr[j];
    Cs[j][tid] = p ? p[tid] : 0.0f;
  }
  __syncthreads();

  const int nl = lane & 15;
  const int kk = (lane >> 4) * 2;
  const int hi = lane >> 4;
  const int n0 = wave * 16;

  v8f acc;
  {
    float bias = down_b[n0 + nl];
    #pragma unroll
    for (int r = 0; r < 8; ++r) acc[r] = bias;
  }
  #pragma unroll 4
  for (int k0 = 0; k0 < CC; k0 += 4) {
    v2f a, b;
    a.x = Cs[nl][k0 + kk];
    a.y = Cs[nl][k0 + kk + 1];
    b.x = down_w[(k0 + kk)     * CF + n0 + nl];
    b.y = down_w[(k0 + kk + 1) * CF + n0 + nl];
    acc = wmma_f32(a, b, acc);
  }
  #pragma unroll
  for (int r = 0; r < 8; ++r)
    Ws[r + 8 * hi][n0 + nl] = acc[r];
  __syncthreads();

  v8f acc2;
  {
    float bias = merge_b[n0 + nl];
    #pragma unroll
    for (int r = 0; r < 8; ++r) acc2[r] = bias;
  }
  const float* mwb = merge_w + CF * CF;
  #pragma unroll 4
  for (int k0 = 0; k0 < CF; k0 += 4) {
    v2f a, b;
    a.x = Ws[nl][k0 + kk];
    a.y = Ws[nl][k0 + kk + 1];
    b.x = mwb[(k0 + kk)     * CF + n0 + nl];
    b.y = mwb[(k0 + kk + 1) * CF + n0 + nl];
    acc2 = wmma_f32(a, b, acc2);
  }
  #pragma unroll
  for (int r = 0; r < 8; ++r) {
    int mm = mm0 + r + 8 * hi;
    if (mm < twoM) t_out[mm * CF + n0 + nl] = acc2[r];
  }
}

extern "C" void kernel_launch(void* const* d_in, const int* in_sizes, int n_in,
                              void* d_out, int out_size, void* d_ws, size_t ws_size,
                              hipStream_t stream) {
  const float* feat_f0 = (const float*)d_in[0];
  const float* feat_f1 = (const float*)d_in[1];
  const float* feat_c0 = (const float*)d_in[2];
  const float* feat_c1 = (const float*)d_in[3];
  const float* down_w  = (const float*)d_in[4];
  const float* down_b  = (const float*)d_in[5];
  const float* merge_w = (const float*)d_in[6];
  const float* merge_b = (const float*)d_in[7];
  const int*   b_ids   = (const int*)d_in[8];
  const int*   i_ids   = (const int*)d_in[9];
  const int*   j_ids   = (const int*)d_in[10];
  // wc/stride scalars are fixed by the reference (80, 8) and hardcoded.

  const int M    = in_sizes[8];
  const int Nimg = in_sizes[0] / (CF * HF * WF);
  const int LC   = in_sizes[2] / (Nimg * CC);

  float* t_ws = (float*)d_ws;   // 2*M*128 floats (~3 MB for M=3000)

  dim3 blk(256);
  int prepBlocks = (2 * M + 15) / 16;
  prep_kernel<<<prepBlocks, blk, 0, stream>>>(
      feat_c0, feat_c1, down_w, down_b, merge_w, merge_b,
      b_ids, i_ids, j_ids, t_ws, M, LC);

  int rows = 2 * M * WW;
  int mainBlocks = (rows + 15) / 16;
  fine_kernel<<<mainBlocks, blk, 0, stream>>>(
      feat_f0, feat_f1, merge_w, b_ids, i_ids, j_ids,
      t_ws, (float*)d_out, M);
}